// LengthRegulator_27023934227157
// MI455X (gfx1250) — compile-verified
//
#include <hip/hip_runtime.h>
#include <stdint.h>

// Problem constants (match reference)
#define BB   32
#define TT   1024
#define DD   384
#define TOUT 4096   // TT * MAX_DUR

// ---------------------------------------------------------------------------
// gfx1250 async global<->LDS data mover (ASYNCcnt-tracked) detection
// ---------------------------------------------------------------------------
#if defined(__AMDGCN__) && \
    __has_builtin(__builtin_amdgcn_global_load_async_to_lds_b128) && \
    __has_builtin(__builtin_amdgcn_global_store_async_from_lds_b128)
#define USE_ASYNC_LDS 1
#else
#define USE_ASYNC_LDS 0
#endif

#if USE_ASYNC_LDS
// Builtin parameter type is a 16-byte int vector pointee.
typedef int v4i_t __attribute__((vector_size(16)));
typedef __attribute__((address_space(1))) v4i_t* gv4i_p;   // global
typedef __attribute__((address_space(3))) v4i_t* lv4i_p;   // LDS

__device__ __forceinline__ gv4i_p as_global_v4(const void* p) {
    // global flat address == global segment address
    return (gv4i_p)(unsigned long long)(uintptr_t)p;
}
__device__ __forceinline__ lv4i_p as_lds_v4(const void* p) {
    // flat LDS address = {SHARED_BASE, offset[31:0]} -> low 32 bits = LDS offset
    return (lv4i_p)(unsigned)(uintptr_t)p;
}
__device__ __forceinline__ void wait_async0() {
#if __has_builtin(__builtin_amdgcn_s_wait_asynccnt)
    __builtin_amdgcn_s_wait_asynccnt(0);
#else
    asm volatile("s_wait_asynccnt 0" ::: "memory");
#endif
}
#endif

// ---------------------------------------------------------------------------
// Kernel 1: masked duration scan per batch.
//   csum[b][t] = inclusive cumsum of (t < len[b] ? dur[b][t] : 0)
//   if total == 0: csum[b][t] = min(t+1, len[b]), total = len[b]
// One 1024-thread block (32 wave32s) per batch; Hillis-Steele scan in LDS.
// ---------------------------------------------------------------------------
__global__ __launch_bounds__(TT) void lr_scan(const int* __restrict__ dur,
                                              const int* __restrict__ lens,
                                              int* __restrict__ csum,
                                              int* __restrict__ total) {
    const int b = blockIdx.x;
    const int t = threadIdx.x;
    const int L = lens[b];

    __shared__ int s[TT];
    s[t] = (t < L) ? dur[b * TT + t] : 0;
    __syncthreads();

#pragma unroll
    for (int off = 1; off < TT; off <<= 1) {
        int v = (t >= off) ? s[t - off] : 0;
        __syncthreads();
        s[t] += v;
        __syncthreads();
    }

    int tot = s[TT - 1];
    int c   = s[t];
    if (tot == 0) {            // reference's d.fill_(valid) fallback
        c   = (t + 1 < L) ? (t + 1) : L;   // min(t+1, L)
        tot = L;
    }
    csum[b * TT + t] = c;
    if (t == 0) total[b] = tot;
}

// ---------------------------------------------------------------------------
// Kernel 2: expand/gather. grid = (TOUT/32, B), block = 256 (8 waves).
// Each wave owns 4 output frames, each with its OWN 1536B LDS slice
// (8 waves * 4 frames * 1536B = 48KB << 320KB/WGP). Pipeline per wave:
//   4x binary search -> issue up to 12 async b128 loads (deep MLP)
//   -> single s_wait_asynccnt 0 -> issue 12 async b128 stores
//   -> fall off the end (S_ENDPGM implicit wait-idle covers the stores).
// Padded frames get zero float4 stores directly. No barriers anywhere.
// ---------------------------------------------------------------------------
__global__ __launch_bounds__(256) void lr_gather(const float* __restrict__ xs,
                                                 const int* __restrict__ csum,
                                                 const int* __restrict__ total,
                                                 float* __restrict__ out) {
    const int b    = blockIdx.y;
    const int lane = threadIdx.x & 31;
    const int wave = threadIdx.x >> 5;

    const int    tot = total[b];
    const int*   cs  = csum + b * TT;
    const float* xb  = xs  + (size_t)b * TT   * DD;
    float*       ob  = out + (size_t)b * TOUT * DD;

    __shared__ __align__(16) float lds[8][4][DD];   // 48 KB: per-wave, per-frame slices

    const int jbase = blockIdx.x * 32 + wave * 4;

    // Phase 0: searchsorted(side='right') for each of the 4 frames (uniform per wave)
    int  srcs[4];
    bool live[4];
#pragma unroll
    for (int k = 0; k < 4; ++k) {
        const int j = jbase + k;
        live[k] = (j < tot);
        int lo = 0, hi = TT;
        if (live[k]) {
            while (lo < hi) {
                int mid = (lo + hi) >> 1;
                if (cs[mid] > j) hi = mid; else lo = mid + 1;
            }
        }
        srcs[k] = (lo < TT) ? lo : (TT - 1);
    }

#if USE_ASYNC_LDS
    // Phase 1: issue all async loads (up to 12 outstanding per wave)
#pragma unroll
    for (int k = 0; k < 4; ++k) {
        if (live[k]) {
            const float* srow  = xb + (size_t)srcs[k] * DD;
            float*       stage = &lds[wave][k][0];
#pragma unroll
            for (int q = 0; q < 3; ++q) {
                const int e = (q * 32 + lane) * 4;           // float offset, 16B granules
                __builtin_amdgcn_global_load_async_to_lds_b128(
                    as_global_v4(srow + e), as_lds_v4(stage + e), 0, 0);
            }
        }
    }
    wait_async0();   // all slices filled; single sync point per wave

    // Phase 2: issue all async stores (drained by S_ENDPGM's implicit wait-idle)
#pragma unroll
    for (int k = 0; k < 4; ++k) {
        float* dst = ob + (size_t)(jbase + k) * DD;
        if (live[k]) {
            float* stage = &lds[wave][k][0];
#pragma unroll
            for (int q = 0; q < 3; ++q) {
                const int e = (q * 32 + lane) * 4;
                __builtin_amdgcn_global_store_async_from_lds_b128(
                    as_global_v4(dst + e), as_lds_v4(stage + e), 0, 0);
            }
        } else {
            const float4 z = make_float4(0.f, 0.f, 0.f, 0.f);
#pragma unroll
            for (int q = 0; q < 3; ++q)
                ((float4*)dst)[q * 32 + lane] = z;
        }
    }
#else
    // Fallback: plain register b128 copy
#pragma unroll
    for (int k = 0; k < 4; ++k) {
        float* dst = ob + (size_t)(jbase + k) * DD;
        if (live[k]) {
            const float* srow = xb + (size_t)srcs[k] * DD;
#pragma unroll
            for (int q = 0; q < 3; ++q)
                ((float4*)dst)[q * 32 + lane] = ((const float4*)srow)[q * 32 + lane];
        } else {
            const float4 z = make_float4(0.f, 0.f, 0.f, 0.f);
#pragma unroll
            for (int q = 0; q < 3; ++q)
                ((float4*)dst)[q * 32 + lane] = z;
        }
    }
#endif
}

// ---------------------------------------------------------------------------
// Launch: inputs in setup_inputs() order: xs(f32), durations(int), lengths(int)
// Workspace: csum[B*T] int32, total[B] int32  (~128.1 KB)
// ---------------------------------------------------------------------------
extern "C" void kernel_launch(void* const* d_in, const int* in_sizes, int n_in,
                              void* d_out, int out_size, void* d_ws, size_t ws_size,
                              hipStream_t stream) {
    (void)in_sizes; (void)n_in; (void)out_size; (void)ws_size;

    const float* xs   = (const float*)d_in[0];
    const int*   dur  = (const int*)d_in[1];
    const int*   lens = (const int*)d_in[2];
    float*       out  = (float*)d_out;

    int* csum  = (int*)d_ws;
    int* total = (int*)((char*)d_ws + (size_t)BB * TT * sizeof(int));

    lr_scan<<<BB, TT, 0, stream>>>(dur, lens, csum, total);

    dim3 grid(TOUT / 32, BB);
    lr_gather<<<grid, 256, 0, stream>>>(xs, csum, total, out);
}